// VGAE_23356032156161
// MI455X (gfx1250) — compile-verified
//
#include <hip/hip_runtime.h>
#include <hip/hip_bf16.h>

typedef float v2f __attribute__((ext_vector_type(2)));
typedef float v8f __attribute__((ext_vector_type(8)));

// ---------------------------------------------------------------------------
// Kernel 0: zero workspace accumulators (Y and Z regions, plus T in between)
// ---------------------------------------------------------------------------
__global__ void vgae_zero_kernel(float4* __restrict__ p, int n4) {
    int t = blockIdx.x * blockDim.x + threadIdx.x;
    if (t < n4) p[t] = make_float4(0.f, 0.f, 0.f, 0.f);
}

// ---------------------------------------------------------------------------
// Kernel 1: H = X @ W1   ([N,512] @ [512,16] -> [N,16])  via v_wmma_f32_16x16x4
// One wave computes one 16x16 output tile. W1 is staged transposed in LDS
// (Wt[n*512+k] = W1[k*16+n]) so each lane's B pair (k,k+1 for fixed n) is a
// contiguous ds_load_b64. A pair (k,k+1 for fixed row) is a global_load_b64.
// ISA 16x4 f32 A layout: lanes 0-15 -> K0,K1 ; lanes 16-31 -> K2,K3 (M=lane%16).
// B (4x16) mirrored: lanes 0-15 -> rows K0,K1 ; lanes 16-31 -> K2,K3 (N=lane%16).
// ---------------------------------------------------------------------------
__global__ void vgae_gemm1_kernel(const float* __restrict__ X,
                                  const float* __restrict__ W1,
                                  float* __restrict__ H,
                                  int n_tiles, int N) {
    __shared__ float Wt[16 * 512];   // 32 KB
    for (int i = threadIdx.x; i < 16 * 512; i += blockDim.x) {
        int n = i >> 9;          // 0..15
        int k = i & 511;         // 0..511
        Wt[i] = W1[k * 16 + n];
    }
    __syncthreads();

    int wave = threadIdx.x >> 5;           // 8 waves / block
    int lane = threadIdx.x & 31;
    int tile = blockIdx.x * 8 + wave;
    if (tile >= n_tiles) return;

    int m    = lane & 15;
    int koff = (lane >> 4) * 2;            // 0 or 2
    int rowA = tile * 16 + m;
    if (rowA >= N) rowA = N - 1;           // clamp (stores are guarded)

    const float* xrow = X  + (size_t)rowA * 512 + koff;
    const float* wrow = Wt + m * 512 + koff;

    v8f c = {0.f, 0.f, 0.f, 0.f, 0.f, 0.f, 0.f, 0.f};
    for (int k = 0; k < 512; k += 4) {
        v2f a = *(const v2f*)(xrow + k);
        v2f b = *(const v2f*)(wrow + k);
        c = __builtin_amdgcn_wmma_f32_16x16x4_f32(
                /*neg_a=*/false, a, /*neg_b=*/false, b,
                /*c_mod=*/(short)0, c, /*reuse_a=*/false, /*reuse_b=*/false);
    }

    // C/D layout: VGPR r, lanes 0-15 -> row tile*16+r ; lanes 16-31 -> +8
    int r0 = tile * 16 + (lane >> 4) * 8;
    int n  = lane & 15;
#pragma unroll
    for (int r = 0; r < 8; ++r) {
        int rw = r0 + r;
        if (rw < N) H[(size_t)rw * 16 + n] = c[r];
    }
}

// ---------------------------------------------------------------------------
// Kernels 2 & 4: COO SpMM with scatter-add.  Thread t handles (edge, feature):
// Acc[row[e]][f] += val[e] * M[col[e]][f].  Feature rows are 16-float (64B)
// so the 16-lane gather is a single coalesced L2-resident line; atomics land
// in L2 (target arrays are ~6.4 MB << 192 MB).
// ---------------------------------------------------------------------------
__global__ void vgae_spmm_kernel(const int*   __restrict__ erow,
                                 const int*   __restrict__ ecol,
                                 const float* __restrict__ eval_,
                                 const float* __restrict__ M,
                                 float* __restrict__ Acc,
                                 long long total, int nf) {
    long long t = (long long)blockIdx.x * blockDim.x + threadIdx.x;
    if (t >= total) return;
    int e = (int)(t >> 4);
    int f = (int)(t & 15);
    if (f >= nf) return;
    int   c = ecol[e];
    int   r = erow[e];
    float v = eval_[e];
    atomicAdd(Acc + (size_t)r * 16 + f, v * M[(size_t)c * 16 + f]);
}

// ---------------------------------------------------------------------------
// Kernel 3: T = relu(Y) @ [W2 | W3 | 0pad]   ([N,16] @ [16,16] -> [N,16])
// Same WMMA tiling as kernel 1; relu fused into the A-operand load. Columns
// 0-6 = W2 product (z_mean pre-SpMM), 7-13 = W3 product, 14-15 zero.
// ---------------------------------------------------------------------------
__global__ void vgae_dense2_kernel(const float* __restrict__ Y,
                                   const float* __restrict__ W2,
                                   const float* __restrict__ W3,
                                   float* __restrict__ T,
                                   int n_tiles, int N) {
    __shared__ float Wt[16 * 16];    // Wt[n*16+k]
    for (int i = threadIdx.x; i < 256; i += blockDim.x) {
        int n = i >> 4;
        int k = i & 15;
        float w = 0.f;
        if (n < 7)       w = W2[k * 7 + n];
        else if (n < 14) w = W3[k * 7 + (n - 7)];
        Wt[i] = w;
    }
    __syncthreads();

    int wave = threadIdx.x >> 5;
    int lane = threadIdx.x & 31;
    int tile = blockIdx.x * 8 + wave;
    if (tile >= n_tiles) return;

    int m    = lane & 15;
    int koff = (lane >> 4) * 2;
    int rowA = tile * 16 + m;
    if (rowA >= N) rowA = N - 1;

    const float* yrow = Y  + (size_t)rowA * 16 + koff;
    const float* wrow = Wt + m * 16 + koff;

    v8f c = {0.f, 0.f, 0.f, 0.f, 0.f, 0.f, 0.f, 0.f};
#pragma unroll
    for (int k = 0; k < 16; k += 4) {
        v2f a = *(const v2f*)(yrow + k);
        a.x = fmaxf(a.x, 0.f);           // fused relu
        a.y = fmaxf(a.y, 0.f);
        v2f b = *(const v2f*)(wrow + k);
        c = __builtin_amdgcn_wmma_f32_16x16x4_f32(
                false, a, false, b, (short)0, c, false, false);
    }

    int r0 = tile * 16 + (lane >> 4) * 8;
    int n  = lane & 15;
#pragma unroll
    for (int r = 0; r < 8; ++r) {
        int rw = r0 + r;
        if (rw < N) T[(size_t)rw * 16 + n] = c[r];
    }
}

// ---------------------------------------------------------------------------
// Kernel 5: out[n][z] = Z[n][z] + eps[n][z] * exp(Z[n][7+z]),  z in 0..6
// ---------------------------------------------------------------------------
__global__ void vgae_finalize_kernel(const float* __restrict__ Z,
                                     const float* __restrict__ eps,
                                     float* __restrict__ out, int total) {
    int t = blockIdx.x * blockDim.x + threadIdx.x;
    if (t >= total) return;
    int node = t / 7;
    int z    = t - node * 7;
    float zm = Z[(size_t)node * 16 + z];
    float zl = Z[(size_t)node * 16 + 7 + z];
    out[t] = zm + eps[t] * __expf(zl);
}

// ---------------------------------------------------------------------------
extern "C" void kernel_launch(void* const* d_in, const int* in_sizes, int n_in,
                              void* d_out, int out_size, void* d_ws, size_t ws_size,
                              hipStream_t stream) {
    const float* X     = (const float*)d_in[0];
    const int*   erow  = (const int*)  d_in[1];
    const int*   ecol  = (const int*)  d_in[2];
    const float* eval_ = (const float*)d_in[3];
    const float* W1    = (const float*)d_in[4];
    const float* W2    = (const float*)d_in[5];
    const float* W3    = (const float*)d_in[6];
    const float* eps   = (const float*)d_in[7];
    float* out = (float*)d_out;

    const int F = 512;
    const int N = in_sizes[0] / F;          // 100000
    const int E = in_sizes[1];              // 3200000
    const int n_tiles = (N + 15) / 16;      // 6250

    // Workspace layout: H | Y | T | Z, each [N,16] f32 (25.6 MB total)
    float* H = (float*)d_ws;
    float* Y = H + (size_t)N * 16;
    float* T = Y + (size_t)N * 16;
    float* Z = T + (size_t)N * 16;

    // Zero Y, T, Z (contiguous 3*N*16 floats starting at Y)
    {
        int n4 = (3 * N * 16) / 4;
        vgae_zero_kernel<<<(n4 + 255) / 256, 256, 0, stream>>>((float4*)Y, n4);
    }

    // 1) H = X @ W1  (WMMA f32 16x16x4)
    vgae_gemm1_kernel<<<(n_tiles + 7) / 8, 256, 0, stream>>>(X, W1, H, n_tiles, N);

    // 2) Y = SpMM(G, H)   (relu deferred to kernel 3)
    {
        long long total = (long long)E * 16;
        int blocks = (int)((total + 255) / 256);
        vgae_spmm_kernel<<<blocks, 256, 0, stream>>>(erow, ecol, eval_, H, Y, total, 16);
    }

    // 3) T = relu(Y) @ [W2|W3]  (WMMA f32 16x16x4)
    vgae_dense2_kernel<<<(n_tiles + 7) / 8, 256, 0, stream>>>(Y, W2, W3, T, n_tiles, N);

    // 4) Z = SpMM(G, T)  -> cols 0-6 = z_mean, cols 7-13 = z_log_std
    {
        long long total = (long long)E * 16;
        int blocks = (int)((total + 255) / 256);
        vgae_spmm_kernel<<<blocks, 256, 0, stream>>>(erow, ecol, eval_, T, Z, total, 14);
    }

    // 5) out = z_mean + eps * exp(z_log_std)
    {
        int total = N * 7;
        vgae_finalize_kernel<<<(total + 255) / 256, 256, 0, stream>>>(Z, eps, out, total);
    }
}